// SuctionPartSampler_34600256536873
// MI455X (gfx1250) — compile-verified
//
#include <hip/hip_runtime.h>
#include <hip/hip_bf16.h>

typedef float v2f __attribute__((ext_vector_type(2)));
typedef float v8f __attribute__((ext_vector_type(8)));

#define LN_EPS 1e-5f
#define NORM_EPS 1e-12f

// D = A(16x4 f32) * B(4x16 f32) + C(16x16 f32), full-precision CDNA5 WMMA.
__device__ __forceinline__ v8f wmma4(v2f a, v2f b, v8f c) {
    return __builtin_amdgcn_wmma_f32_16x16x4_f32(
        /*neg_a=*/false, a, /*neg_b=*/false, b,
        /*c_mod=*/(short)0, c, /*reuse_a=*/false, /*reuse_b=*/false);
}

__global__ __launch_bounds__(256) void suction_decoder_kernel(
    const float* __restrict__ x,   // [N,64]
    const float* __restrict__ w1,  // [64,32]
    const float* __restrict__ g1, const float* __restrict__ b1,   // [32]
    const float* __restrict__ w2,  // [32,16]
    const float* __restrict__ g2, const float* __restrict__ b2,   // [16]
    const float* __restrict__ w3,  // [16,3]
    const float* __restrict__ b3,  // [3]
    const int* __restrict__ pH, const int* __restrict__ pW,
    float* __restrict__ out,       // [B,3,H,W] flat
    int N)
{
    const int lane = threadIdx.x & 31;
    const int wid  = threadIdx.x >> 5;
    const int half = lane >> 4;       // which 16-lane half
    const int lm   = lane & 15;       // lane mod 16

    const unsigned HW = (unsigned)((*pH) * (*pW));

    // ---- Preload B matrices in WMMA B layout ----
    // B 4x16: VGPR0 = {K=0 (lanes 0-15), K=2 (lanes 16-31)}, VGPR1 = {K=1, K=3}; lane%16 = N.
    v2f Bw1[2][16];   // layer1: K=64 -> 16 k-steps, two 16-col tiles (N=0..15, 16..31)
    #pragma unroll
    for (int kk = 0; kk < 16; ++kk) {
        const int k0 = 4 * kk + 2 * half;
        #pragma unroll
        for (int t = 0; t < 2; ++t) {
            const int n = lm + 16 * t;
            v2f b; b.x = w1[k0 * 32 + n]; b.y = w1[(k0 + 1) * 32 + n];
            Bw1[t][kk] = b;
        }
    }
    v2f Bw2[8];       // layer2: K=32 -> 8 k-steps, one 16-col tile
    #pragma unroll
    for (int kk = 0; kk < 8; ++kk) {
        const int k0 = 4 * kk + 2 * half;
        v2f b; b.x = w2[k0 * 16 + lm]; b.y = w2[(k0 + 1) * 16 + lm];
        Bw2[kk] = b;
    }
    v2f Bw3[4];       // layer3: K=16 -> 4 k-steps; w3 zero-padded to 16 output cols
    #pragma unroll
    for (int kk = 0; kk < 4; ++kk) {
        const int k0 = 4 * kk + 2 * half;
        v2f b;
        b.x = (lm < 3) ? w3[k0 * 3 + lm] : 0.0f;
        b.y = (lm < 3) ? w3[(k0 + 1) * 3 + lm] : 0.0f;
        Bw3[kk] = b;
    }
    // gamma/beta in A-layout K-pairs (per-lane feature pair per k-step)
    v2f Gw1[8], Bq1[8];
    #pragma unroll
    for (int kk = 0; kk < 8; ++kk) {
        const int k0 = 4 * kk + 2 * half;
        Gw1[kk].x = g1[k0]; Gw1[kk].y = g1[k0 + 1];
        Bq1[kk].x = b1[k0]; Bq1[kk].y = b1[k0 + 1];
    }
    v2f Gw2[4], Bq2[4];
    #pragma unroll
    for (int kk = 0; kk < 4; ++kk) {
        const int k0 = 4 * kk + 2 * half;
        Gw2[kk].x = g2[k0]; Gw2[kk].y = g2[k0 + 1];
        Bq2[kk].x = b2[k0]; Bq2[kk].y = b2[k0 + 1];
    }
    const float b3v = (lm < 3) ? b3[lm] : 0.0f;   // padded channels stay exactly 0

    // Wave-private LDS transpose buffer: 16 rows x 32 features per wave.
    __shared__ float lds[8][16][32];

    const int numTiles = N >> 4;
    for (int tile = blockIdx.x * 8 + wid; tile < numTiles; tile += gridDim.x * 8) {
        const int row0 = tile << 4;

        // ---- Load A tile (16x64) in WMMA A layout: lane%16 = row, v2f = K pair ----
        const float* xrow = x + (size_t)(row0 + lm) * 64 + 2 * half;
        v2f A[16];
        #pragma unroll
        for (int kk = 0; kk < 16; ++kk)
            A[kk] = *(const v2f*)(xrow + 4 * kk);

        // ---- Layer 1: [16x64] @ [64x32] via 32 fp32 WMMAs ----
        v8f C0 = {}, C1 = {};
        #pragma unroll
        for (int kk = 0; kk < 16; ++kk) {
            C0 = wmma4(A[kk], Bw1[0][kk], C0);
            C1 = wmma4(A[kk], Bw1[1][kk], C1);
        }

        // ---- Transpose raw GEMM output through LDS into A layout ----
        // C layout: lane%16 = feature (tile0: 0-15, tile1: 16-31), VGPR v = row v + 8*half.
        #pragma unroll
        for (int v = 0; v < 8; ++v) {
            const int row = v + 8 * half;
            lds[wid][row][lm]      = C0[v];
            lds[wid][row][lm + 16] = C1[v];
        }
        asm volatile("s_wait_dscnt 0x0" ::: "memory");
        v2f A2[8];                         // row lm, feature pair {4kk+2h, +1}
        #pragma unroll
        for (int kk = 0; kk < 8; ++kk)
            A2[kk] = *(const v2f*)&lds[wid][lm][4 * kk + 2 * half];

        // ---- LayerNorm(32)+ReLU in A layout: row split across lanes lm / lm+16 ----
        {
            float s = 0.0f, sq = 0.0f;
            #pragma unroll
            for (int kk = 0; kk < 8; ++kk) {
                s += A2[kk].x + A2[kk].y;
                sq = fmaf(A2[kk].x, A2[kk].x, fmaf(A2[kk].y, A2[kk].y, sq));
            }
            s  += __shfl_xor(s, 16, 32);   // combine the two half-lanes of this row
            sq += __shfl_xor(sq, 16, 32);
            const float mu  = s * (1.0f / 32.0f);
            const float var = sq * (1.0f / 32.0f) - mu * mu;
            const float rs  = rsqrtf(var + LN_EPS);
            #pragma unroll
            for (int kk = 0; kk < 8; ++kk) {
                A2[kk].x = fmaxf((A2[kk].x - mu) * rs * Gw1[kk].x + Bq1[kk].x, 0.0f);
                A2[kk].y = fmaxf((A2[kk].y - mu) * rs * Gw1[kk].y + Bq1[kk].y, 0.0f);
            }
        }

        // ---- Layer 2: [16x32] @ [32x16] via 8 fp32 WMMAs ----
        v8f C2 = {};
        #pragma unroll
        for (int kk = 0; kk < 8; ++kk)
            C2 = wmma4(A2[kk], Bw2[kk], C2);

        // ---- Transpose raw layer-2 output, LN(16)+ReLU in A layout ----
        #pragma unroll
        for (int v = 0; v < 8; ++v)
            lds[wid][v + 8 * half][lm] = C2[v];
        asm volatile("s_wait_dscnt 0x0" ::: "memory");
        v2f A3[4];
        #pragma unroll
        for (int kk = 0; kk < 4; ++kk)
            A3[kk] = *(const v2f*)&lds[wid][lm][4 * kk + 2 * half];
        {
            float s = 0.0f, sq = 0.0f;
            #pragma unroll
            for (int kk = 0; kk < 4; ++kk) {
                s += A3[kk].x + A3[kk].y;
                sq = fmaf(A3[kk].x, A3[kk].x, fmaf(A3[kk].y, A3[kk].y, sq));
            }
            s  += __shfl_xor(s, 16, 32);
            sq += __shfl_xor(sq, 16, 32);
            const float mu  = s * (1.0f / 16.0f);
            const float var = sq * (1.0f / 16.0f) - mu * mu;
            const float rs  = rsqrtf(var + LN_EPS);
            #pragma unroll
            for (int kk = 0; kk < 4; ++kk) {
                A3[kk].x = fmaxf((A3[kk].x - mu) * rs * Gw2[kk].x + Bq2[kk].x, 0.0f);
                A3[kk].y = fmaxf((A3[kk].y - mu) * rs * Gw2[kk].y + Bq2[kk].y, 0.0f);
            }
        }

        // ---- Layer 3: [16x16] @ [16x16-padded] via 4 fp32 WMMAs ----
        v8f C3 = {};
        #pragma unroll
        for (int kk = 0; kk < 4; ++kk)
            C3 = wmma4(A3[kk], Bw3[kk], C3);

        // ---- +b3, channel L2-normalize, scatter store to [B,3,H,W] ----
        // C3: lane%16 = channel (0..2 valid, rest exactly 0), VGPR v = row v + 8*half.
        unsigned rowL = (unsigned)(row0 + 8 * half);   // first row this half handles
        unsigned bimg = rowL / HW;                     // one division per tile
        unsigned rem  = rowL - bimg * HW;
        #pragma unroll
        for (int v = 0; v < 8; ++v) {
            const float o  = C3[v] + b3v;              // zero for channels >= 3
            float sq = o * o;
            sq += __shfl_xor(sq, 1, 32);               // sum over the 3 channels
            sq += __shfl_xor(sq, 2, 32);               // (lane 3 contributes 0)
            const float inv = 1.0f / fmaxf(sqrtf(sq), NORM_EPS);
            if (lm < 3)
                out[(size_t)(bimg * 3u + (unsigned)lm) * HW + rem] = o * inv;
            // advance to next row (carry into next image if needed)
            rem += 1u;
            if (rem == HW) { rem = 0u; bimg += 1u; }
        }
    }
}

extern "C" void kernel_launch(void* const* d_in, const int* in_sizes, int n_in,
                              void* d_out, int out_size, void* d_ws, size_t ws_size,
                              hipStream_t stream) {
    const float* x  = (const float*)d_in[0];
    const float* w1 = (const float*)d_in[1];
    const float* g1 = (const float*)d_in[2];
    const float* b1 = (const float*)d_in[3];
    const float* w2 = (const float*)d_in[4];
    const float* g2 = (const float*)d_in[5];
    const float* b2 = (const float*)d_in[6];
    const float* w3 = (const float*)d_in[7];
    const float* b3 = (const float*)d_in[8];
    const int* pH = (const int*)d_in[10];
    const int* pW = (const int*)d_in[11];
    float* out = (float*)d_out;

    const int N = in_sizes[0] / 64;          // rows
    const int numTiles = N / 16;             // 16-row tiles, one per wave iteration
    const int tilesPerWave = 8;              // amortize weight preload over 8 tiles
    int blocks = (numTiles + 8 * tilesPerWave - 1) / (8 * tilesPerWave);
    if (blocks < 1) blocks = 1;

    suction_decoder_kernel<<<blocks, 256, 0, stream>>>(
        x, w1, g1, b1, w2, g2, b2, w3, b3, pH, pW, out, N);
}